// DiscriminativeLoss_51453708206467
// MI455X (gfx1250) — compile-verified
//
#include <hip/hip_runtime.h>
#include <math.h>

// ---------------- problem constants ----------------
#define Bn 8
#define En 16
#define Hn 512
#define Wn 512
#define Nn (Hn * Wn)   // 262144 pixels per batch image
#define Kn 32          // instance ids 1..K, 0 = background

// DELTA_VAR = 0.5, DELTA_DIST = 1.5 (2*DELTA_DIST = 3.0)
// ALPHA = BETA = 1.0, GAMMA = 0.001

typedef __attribute__((ext_vector_type(2))) float v2f;
typedef __attribute__((ext_vector_type(8))) float v8f;

// ---------------- workspace layout (floats) ----------------
#define OFF_SUM 0                         // [B][K][E] center accumulators
#define OFF_CNT (OFF_SUM + Bn * Kn * En)  // [B][K]    pixel counts (as float)
#define OFF_CEN (OFF_CNT + Bn * Kn)       // [B][K][E] finalized centers
#define OFF_DST (OFF_CEN + Bn * Kn * En)  // [B]       loss_dist_b
#define OFF_REG (OFF_DST + Bn)            // [B]       loss_reg_b
#define OFF_NIN (OFF_REG + Bn)            // [B]       n_inst
#define OFF_PIN (OFF_NIN + Bn)            // [B][K]    sum of hinged^2 per instance
#define WS_FLOATS (OFF_PIN + Bn * Kn)     // ~8.7k floats (~35 KB)

// ---------------- kernel 0: zero the accumulators ----------------
__global__ __launch_bounds__(256) void k_zero(float* __restrict__ w) {
    int i = blockIdx.x * 256 + threadIdx.x;
    if (i < WS_FLOATS) w[i] = 0.0f;
}

// ---------------- kernel 1: centers_sum via V_WMMA_F32_16X16X4_F32 ----------------
// D[e,k] += A[e,p] * B[p,k], A = fp32 embedding tile (16 rows = all of E,
// 4 pixels), B = exact one-hot built from the instance mask in registers.
// Each wave handles 512 pixels; each block (8 waves) handles 4096 pixels.
__global__ __launch_bounds__(256) void k_centers(const float* __restrict__ emb,
                                                 const int* __restrict__ mask,
                                                 float* __restrict__ w) {
    __shared__ float ls[Kn * En];
    __shared__ int lc[Kn];

    const int b = blockIdx.y;
    const int tid = threadIdx.x;
    for (int i = tid; i < Kn * En; i += 256) ls[i] = 0.0f;
    if (tid < Kn) lc[tid] = 0;
    __syncthreads();

    const int lane = tid & 31;
    const int half = lane >> 4;   // 0: lanes 0-15, 1: lanes 16-31
    const int col  = lane & 15;   // A row (e) / B,D column (k)
    const int wave = tid >> 5;
    const int wbase = blockIdx.x * 4096 + wave * 512;

    const float* __restrict__ embb = emb + (size_t)b * En * Nn;
    const int*   __restrict__ mb   = mask + (size_t)b * Nn;
    const float* __restrict__ arow = embb + (size_t)col * Nn;

    v8f accLo = {};  // D for instances k = 0..15
    v8f accHi = {};  // D for instances k = 16..31

    for (int s = 0; s < 128; ++s) {
        // ISA 16x4 fp32 A layout: VGPR0 holds K={0 | 2}, VGPR1 holds K={1 | 3}
        // across the two lane halves -> a single b64 load lands in-layout.
        const int p = wbase + s * 4 + half * 2;
        v2f a = *(const v2f*)(arow + p);

        const int m0 = mb[p];
        const int m1 = mb[p + 1];
        v2f blo, bhi;
        blo.x = (m0 == col + 1)  ? 1.0f : 0.0f;  // B row K = half*2,   col k = col
        blo.y = (m1 == col + 1)  ? 1.0f : 0.0f;  // B row K = half*2+1
        bhi.x = (m0 == col + 17) ? 1.0f : 0.0f;  // same rows, k = col+16
        bhi.y = (m1 == col + 17) ? 1.0f : 0.0f;

        accLo = __builtin_amdgcn_wmma_f32_16x16x4_f32(
            false, a, false, blo, (short)0, accLo, false, false);
        accHi = __builtin_amdgcn_wmma_f32_16x16x4_f32(
            false, a, false, bhi, (short)0, accHi, false, false);
    }

    // per-instance pixel counts for this wave's 512 pixels
    for (int p = lane; p < 512; p += 32) {
        int m = mb[wbase + p];
        if (m > 0) atomicAdd(&lc[m - 1], 1);
    }

    // D element at (lane, vgpr v) is (M = v + 8*half, N = col)
#pragma unroll
    for (int v = 0; v < 8; ++v) {
        const int e = v + 8 * half;
        atomicAdd(&ls[col * En + e], accLo[v]);
        atomicAdd(&ls[(col + 16) * En + e], accHi[v]);
    }
    __syncthreads();

    float* __restrict__ sum = w + OFF_SUM + (size_t)b * Kn * En;
    for (int i = tid; i < Kn * En; i += 256) atomicAdd(&sum[i], ls[i]);
    if (tid < Kn) atomicAdd(&w[OFF_CNT + b * Kn + tid], (float)lc[tid]);
}

// ---------------- kernel 2: finalize centers, dist & reg terms ----------------
__global__ __launch_bounds__(256) void k_finalize(float* __restrict__ w) {
    const int b = blockIdx.x;
    const int tid = threadIdx.x;
    __shared__ float c[Kn * En];
    __shared__ float cnts[Kn];
    __shared__ float red[3];  // 0: hinged-pair sum, 1: reg sum, 2: n_inst

    if (tid < Kn) cnts[tid] = w[OFF_CNT + b * Kn + tid];
    if (tid < 3) red[tid] = 0.0f;
    __syncthreads();

    for (int i = tid; i < Kn * En; i += 256) {
        const int k = i / En;
        const float cc = w[OFF_SUM + b * Kn * En + i] / fmaxf(cnts[k], 1.0f);
        c[i] = cc;
        w[OFF_CEN + b * Kn * En + i] = cc;
    }
    __syncthreads();

    if (tid < Kn && cnts[tid] > 0.0f) {
        float csq = 0.0f;
#pragma unroll
        for (int e = 0; e < En; ++e) { float t = c[tid * En + e]; csq += t * t; }
        atomicAdd(&red[1], sqrtf(csq));
        atomicAdd(&red[2], 1.0f);
    }
    __syncthreads();

    for (int idx = tid; idx < Kn * Kn; idx += 256) {
        const int i = idx >> 5, j = idx & 31;
        if (i < j && cnts[i] > 0.0f && cnts[j] > 0.0f) {
            float dsq = 0.0f;
#pragma unroll
            for (int e = 0; e < En; ++e) {
                float t = c[i * En + e] - c[j * En + e];
                dsq += t * t;
            }
            float h = fmaxf(3.0f - sqrtf(dsq), 0.0f);  // 2*DELTA_DIST = 3.0
            atomicAdd(&red[0], h * h);
        }
    }
    __syncthreads();

    if (tid == 0) {
        const float n = red[2];
        const float npairs = n * (n - 1.0f) * 0.5f;
        w[OFF_DST + b] = red[0] / fmaxf(npairs, 1.0f);
        w[OFF_REG + b] = red[1] / fmaxf(n, 1.0f);
        w[OFF_NIN + b] = n;
    }
}

// ---------------- kernel 3: variance term (second coalesced pass) ----------------
// For fixed e, consecutive lanes read consecutive pixels -> fully coalesced.
__global__ __launch_bounds__(256) void k_var(const float* __restrict__ emb,
                                             const int* __restrict__ mask,
                                             float* __restrict__ w) {
    __shared__ float c[Kn * En];
    __shared__ float hs[Kn];
    const int b = blockIdx.y;
    const int tid = threadIdx.x;

    for (int i = tid; i < Kn * En; i += 256) c[i] = w[OFF_CEN + b * Kn * En + i];
    if (tid < Kn) hs[tid] = 0.0f;
    __syncthreads();

    const float* __restrict__ embb = emb + (size_t)b * En * Nn;
    const int*   __restrict__ mb   = mask + (size_t)b * Nn;
    const int base = blockIdx.x * 4096;

    for (int p = base + tid; p < base + 4096; p += 256) {
        const int m = mb[p];
        if (m > 0) {
            const float* ce = &c[(m - 1) * En];
            float d2 = 0.0f;
#pragma unroll
            for (int e = 0; e < En; ++e) {
                float t = embb[(size_t)e * Nn + p] - ce[e];
                d2 += t * t;
            }
            float h = fmaxf(sqrtf(d2) - 0.5f, 0.0f);  // DELTA_VAR = 0.5
            atomicAdd(&hs[m - 1], h * h);
        }
    }
    __syncthreads();
    if (tid < Kn) atomicAdd(&w[OFF_PIN + b * Kn + tid], hs[tid]);
}

// ---------------- kernel 4: final 4-scalar reduction ----------------
__global__ __launch_bounds__(64) void k_final(const float* __restrict__ w,
                                              float* __restrict__ out) {
    __shared__ float varb[Bn];
    const int tid = threadIdx.x;
    if (tid < Bn) {
        float s = 0.0f;
        for (int k = 0; k < Kn; ++k)
            s += w[OFF_PIN + tid * Kn + k] / fmaxf(w[OFF_CNT + tid * Kn + k], 1.0f);
        varb[tid] = s / fmaxf(w[OFF_NIN + tid], 1.0f);
    }
    __syncthreads();
    if (tid == 0) {
        float vb = 0.0f, Lv = 0.0f, Ld = 0.0f, Lr = 0.0f;
        for (int b = 0; b < Bn; ++b) {
            const float v = (w[OFF_NIN + b] > 0.0f) ? 1.0f : 0.0f;
            vb += v;
            Lv += varb[b] * v;
            Ld += w[OFF_DST + b] * v;
            Lr += w[OFF_REG + b] * v;
        }
        const float d = fmaxf(vb, 1.0f);
        Lv /= d; Ld /= d; Lr /= d;
        out[0] = Lv + Ld + 0.001f * Lr;  // ALPHA*Lv + BETA*Ld + GAMMA*Lr
        out[1] = Lv;
        out[2] = Ld;
        out[3] = Lr;
    }
}

// ---------------- launcher ----------------
extern "C" void kernel_launch(void* const* d_in, const int* in_sizes, int n_in,
                              void* d_out, int out_size, void* d_ws, size_t ws_size,
                              hipStream_t stream) {
    (void)in_sizes; (void)n_in; (void)out_size; (void)ws_size;
    const float* emb  = (const float*)d_in[0];  // [B,E,H,W] fp32
    const int*   mask = (const int*)d_in[1];    // [B,H,W] int32
    float* w   = (float*)d_ws;
    float* out = (float*)d_out;                 // (total, L_var, L_dist, L_reg)

    k_zero<<<(WS_FLOATS + 255) / 256, 256, 0, stream>>>(w);
    k_centers<<<dim3(Nn / 4096, Bn), 256, 0, stream>>>(emb, mask, w);
    k_finalize<<<Bn, 256, 0, stream>>>(w);
    k_var<<<dim3(Nn / 4096, Bn), 256, 0, stream>>>(emb, mask, w);
    k_final<<<1, 64, 0, stream>>>(w, out);
}